// base_sampler_pop_60541859004801
// MI455X (gfx1250) — compile-verified
//
#include <hip/hip_runtime.h>
#include <math.h>
#include <stdint.h>

// ---------------------------------------------------------------------------
// Gumbel top-k negative sampler for MI455X (gfx1250).
//  B = 2048 rows, N = 100000 items, K = 64.
//  Roofline: u is ~819MB read exactly once -> ~35us at 23.3 TB/s. To stay
//  memory-bound, the hot loop must be nearly free of transcendentals:
//    key > thr  ==>  gumbel(u) > thr - lp_max  <=>  u > u_cut,
//    u_cut = exp(-exp(-(thr - lp_max - margin)))   (scalar, updated rarely)
//  so 99%+ of elements cost one compare. Survivors take a slow path that
//  gathers pop[j] from L2 and computes the exact key.
//  Data movement: u streamed through LDS with CDNA5 async loads
//  (global_load_async_to_lds_b128), 4 buffers, 3 tiles in flight,
//  synchronized with s_wait_asynccnt + workgroup barriers.
// ---------------------------------------------------------------------------

#define NT    256      // threads per block (8 waves of 32)
#define K     64       // top-k
#define TILE  1024     // floats per tile (4 KB) => one B128 per lane per tile
#define NBUF  4        // LDS buffers (3 tiles in flight)
#define POOL  4096     // candidate pool entries (power of 2 for bitonic)

// ---- packed (key, index), ordered-float encoding ------------------------
// Descending u64 order == descending key order; low bits hold ~idx so ties
// prefer the smaller index (matches jax.lax.top_k).
__device__ __forceinline__ unsigned long long pack_ki(float k, unsigned idx) {
  unsigned b = __float_as_uint(k);
  b = (b & 0x80000000u) ? ~b : (b | 0x80000000u);
  return ((unsigned long long)b << 32) | (unsigned long long)(~idx);
}
__device__ __forceinline__ float unpack_key(unsigned long long p) {
  unsigned b = (unsigned)(p >> 32);
  b = (b & 0x80000000u) ? (b & 0x7FFFFFFFu) : ~b;
  return __uint_as_float(b);
}
__device__ __forceinline__ int unpack_idx(unsigned long long p) {
  return (int)(~(unsigned)p);
}

// ---- CDNA5 async global->LDS copy (B128: 16 bytes per lane) -------------
__device__ __forceinline__ void async_b128(unsigned lds_off, unsigned long long gaddr) {
  asm volatile("global_load_async_to_lds_b128 %0, %1, off"
               :: "v"(lds_off), "v"(gaddr) : "memory");
}
__device__ __forceinline__ void wait_async_le3() {
  asm volatile("s_wait_asynccnt 3" ::: "memory");
}
__device__ __forceinline__ void wait_async_le0() {
  asm volatile("s_wait_asynccnt 0" ::: "memory");
}

// ---- exact key for item j given u value ---------------------------------
__device__ __forceinline__ float exact_key(float uu, float pp, float lS) {
  float lp = __logf(__logf(pp + 1.0f)) - lS;          // logp[j]
  float uc = fminf(fmaxf(uu, 1e-10f), 1.0f - 1e-10f);
  return lp - __logf(-__logf(uc));                    // logp + gumbel
}

// ---- prep kernel 1: per-block partial sum & max of log(pop+1) -----------
__global__ void logc_partial_kernel(const float* __restrict__ pop,
                                    float* __restrict__ psum,
                                    float* __restrict__ pmax, int n) {
  __shared__ float shs[NT];
  __shared__ float shm[NT];
  int tid = threadIdx.x;
  int i = blockIdx.x * NT + tid;
  float v = (i < n) ? __logf(pop[i] + 1.0f) : 0.0f;
  float m = (i < n) ? v : -3.4e38f;
  shs[tid] = v; shm[tid] = m;
  __syncthreads();
  for (int s = NT >> 1; s > 0; s >>= 1) {
    if (tid < s) { shs[tid] += shs[tid + s]; shm[tid] = fmaxf(shm[tid], shm[tid + s]); }
    __syncthreads();
  }
  if (tid == 0) { psum[blockIdx.x] = shs[0]; pmax[blockIdx.x] = shm[0]; }
}

// ---- prep kernel 2: scalars[0]=logS, scalars[1]=lp_max ------------------
__global__ void finalize_kernel(const float* __restrict__ psum,
                                const float* __restrict__ pmax,
                                int nparts, float* __restrict__ scalars) {
  __shared__ float shs[NT];
  __shared__ float shm[NT];
  int tid = threadIdx.x;
  float acc = 0.0f, m = -3.4e38f;
  for (int i = tid; i < nparts; i += NT) { acc += psum[i]; m = fmaxf(m, pmax[i]); }
  shs[tid] = acc; shm[tid] = m;
  __syncthreads();
  for (int s = NT >> 1; s > 0; s >>= 1) {
    if (tid < s) { shs[tid] += shs[tid + s]; shm[tid] = fmaxf(shm[tid], shm[tid + s]); }
    __syncthreads();
  }
  if (tid == 0) {
    float logS = __logf(shs[0]);
    scalars[0] = logS;
    scalars[1] = __logf(shm[0]) - logS;   // lp_max
  }
}

// ---- block-wide descending bitonic sort of the pool ---------------------
__device__ void sort_pool(unsigned long long* pool, int c, int tid) {
  for (int i = c + tid; i < POOL; i += NT) pool[i] = 0ull;   // pad (= -inf)
  for (unsigned kk = 2; kk <= POOL; kk <<= 1) {
    for (unsigned jj = kk >> 1; jj > 0; jj >>= 1) {
      __syncthreads();
      for (unsigned i = tid; i < POOL / 2; i += NT) {
        unsigned ix = ((i & ~(jj - 1)) << 1) | (i & (jj - 1));
        unsigned jx = ix | jj;
        unsigned long long a = pool[ix], b = pool[jx];
        bool desc = ((ix & kk) == 0);
        if (desc ? (a < b) : (a > b)) { pool[ix] = b; pool[jx] = a; }
      }
    }
  }
  __syncthreads();
}

// ---- main kernel: one workgroup per row ---------------------------------
__global__ void __launch_bounds__(NT)
gumbel_topk_kernel(const float* __restrict__ u,
                   const float* __restrict__ pop,
                   const float* __restrict__ scalars,
                   float* __restrict__ out,
                   int n, int batch) {
  __shared__ float sh_u[NBUF][TILE];
  __shared__ unsigned long long pool[POOL];
  __shared__ int cnt;
  __shared__ float thr;
  __shared__ float ucut;

  const int tid = threadIdx.x;
  const int row = blockIdx.x;
  const float lS = scalars[0];
  const float lp_max = scalars[1];
  const float* urow = u + (size_t)row * (size_t)n;

  if (tid == 0) { cnt = 0; thr = -3.4e38f; ucut = 0.0f; }
  __syncthreads();

  const int ntiles = n / TILE;           // full tiles via async LDS path
  const int tail_base = ntiles * TILE;   // remainder via direct loads

  // ---- pre-issue tiles 0..2 (one B128 per lane per tile) ----
  for (int tt = 0; tt < 3 && tt < ntiles; ++tt) {
    unsigned lu = (unsigned)(uintptr_t)&sh_u[tt & (NBUF - 1)][0] + tid * 16u;
    unsigned long long au =
        (unsigned long long)(uintptr_t)(urow + (size_t)tt * TILE) + tid * 16u;
    async_b128(lu, au);
  }

  for (int t = 0; t < ntiles; ++t) {
    if (t + 3 < ntiles) {
      const int ob = (t + 3) & (NBUF - 1);
      unsigned lu = (unsigned)(uintptr_t)&sh_u[ob][0] + tid * 16u;
      unsigned long long au =
          (unsigned long long)(uintptr_t)(urow + (size_t)(t + 3) * TILE) + tid * 16u;
      async_b128(lu, au);
      wait_async_le3();        // oldest in-flight tile (t) complete, 3 pending
    } else {
      wait_async_le0();        // drain at the pipeline tail
    }
    __syncthreads();           // all waves' async data visible in LDS

    const float thr_l = thr;
    const float ucut_l = ucut;
    const int j0 = t * TILE + tid * 4;

    // one float4 per thread per tile; fast path = 3 max + 1 compare
    const float4 uv = *(const float4*)&sh_u[t & (NBUF - 1)][tid * 4];
    float mx = fmaxf(fmaxf(uv.x, uv.y), fmaxf(uv.z, uv.w));
    float mxc = fminf(fmaxf(mx, 1e-10f), 1.0f - 1e-10f);
    if (mxc > ucut_l) {
      float uu[4] = {uv.x, uv.y, uv.z, uv.w};
      #pragma unroll
      for (int e = 0; e < 4; ++e) {
        float uc = fminf(fmaxf(uu[e], 1e-10f), 1.0f - 1e-10f);
        if (uc > ucut_l) {     // could beat thr only if this holds
          float key = exact_key(uu[e], pop[j0 + e], lS);
          if (key > thr_l) {
            int p = atomicAdd(&cnt, 1);
            if (p < POOL) pool[p] = pack_ki(key, (unsigned)(j0 + e));
          }
        }
      }
    }
    __syncthreads();           // pushes visible; tile buffer free for reuse

    int c = cnt;               // uniform after barrier
    if (c > POOL - TILE) {     // compact: keep top-K, raise thr & ucut
      sort_pool(pool, c, tid);
      if (tid == 0) {
        int keep = (c < K) ? c : K;
        cnt = keep;
        if (keep == K) {
          float t64 = unpack_key(pool[K - 1]);
          thr = t64;
          // conservative filter threshold (margin guards fp rounding)
          float d = t64 - lp_max - 0.01f;
          ucut = __expf(-__expf(-d));
        }
      }
      __syncthreads();
    }
  }

  // ---- tail (n % TILE): direct coalesced global loads ----
  {
    const float thr_l = thr;
    const float ucut_l = ucut;
    for (int j = tail_base + tid; j < n; j += NT) {
      float uu = urow[j];
      float uc = fminf(fmaxf(uu, 1e-10f), 1.0f - 1e-10f);
      if (uc > ucut_l) {
        float key = exact_key(uu, pop[j], lS);
        if (key > thr_l) {
          int p = atomicAdd(&cnt, 1);
          if (p < POOL) pool[p] = pack_ki(key, (unsigned)j);
        }
      }
    }
  }
  __syncthreads();

  // ---- final descending sort: top-K at pool[0..K) ----
  sort_pool(pool, cnt, tid);

  // ---- outputs: items (as float) then items_prob, concatenated flat ----
  if (tid < K) {
    unsigned long long pk = pool[tid];
    int idx = unpack_idx(pk);
    float lc = __logf(pop[idx] + 1.0f);
    float lp = __logf(lc) - lS;
    out[(size_t)row * K + tid] = (float)idx;                  // items
    out[(size_t)batch * K + (size_t)row * K + tid] = lp;      // items_prob
  }
}

// ---------------------------------------------------------------------------
extern "C" void kernel_launch(void* const* d_in, const int* in_sizes, int n_in,
                              void* d_out, int out_size, void* d_ws, size_t ws_size,
                              hipStream_t stream) {
  // inputs: [0] user_id (unused), [1] pop_count f32[N], [2] u f32[B,N],
  //         [3] num_neg (fixed 64)
  const float* pop = (const float*)d_in[1];
  const float* u   = (const float*)d_in[2];
  float* out = (float*)d_out;

  const int n     = in_sizes[1];   // 100000
  const int batch = in_sizes[0];   // 2048

  // tiny scratch: partial sums, partial maxes, 2 scalars (logS, lp_max)
  const int nparts = (n + NT - 1) / NT;
  float* psum    = (float*)d_ws;
  float* pmax    = psum + nparts;
  float* scalars = pmax + nparts;

  logc_partial_kernel<<<nparts, NT, 0, stream>>>(pop, psum, pmax, n);
  finalize_kernel<<<1, NT, 0, stream>>>(psum, pmax, nparts, scalars);
  gumbel_topk_kernel<<<batch, NT, 0, stream>>>(u, pop, scalars, out, n, batch);
}